// ECTLayer_29429115912774
// MI455X (gfx1250) — compile-verified
//
#include <hip/hip_runtime.h>
#include <hip/hip_bf16.h>

typedef float v2f __attribute__((ext_vector_type(2)));
typedef float v8f __attribute__((ext_vector_type(8)));

constexpr int   T_DIRS   = 32;
constexpr int   S_STEPS  = 32;
constexpr int   N_GRAPHS = 64;
constexpr float RADIUS_F = 1.1f;
constexpr float SCALE_F  = 100.0f;

// ---------------------------------------------------------------------------
// Kernel 1: nh[n, t] = (x[n,:] * w[n]) @ v[:, t]   via V_WMMA_F32_16X16X4_F32
// One wave per 16-node tile. A = 16x4 f32 (K=3 zero-padded), B = 4x16 f32
// (two tiles: t=0..15 and t=16..31), C/D = 16x16 f32.
// A layout (ISA 7.12.2): lanes 0-15 -> M=0..15 {K0,K1}; lanes 16-31 -> {K2,K3}
// D layout: lanes 0-15 -> N=lane, M=r; lanes 16-31 -> N=lane-16, M=r+8
// OOB tail rows get w=0 -> zero A row -> zero D row; stores guarded only in
// the (never-taken for N%16==0) tail tile.
// ---------------------------------------------------------------------------
__global__ __launch_bounds__(256)
void nh_wmma_kernel(const float* __restrict__ x,
                    const float* __restrict__ nw,
                    const float* __restrict__ v,
                    float* __restrict__ nh, int N) {
    int wave = (int)((blockIdx.x * (unsigned)blockDim.x + threadIdx.x) >> 5);
    int lane = threadIdx.x & 31;
    int numTiles = (N + 15) >> 4;
    if (wave >= numTiles) return;          // wave-uniform: EXEC stays all-ones

    const int  n0   = wave * 16;
    const int  tl   = lane & 15;
    const bool hi   = lane >= 16;
    const int  node = n0 + tl;
    const int  nc   = node < N ? node : N - 1;   // clamped load index
    const bool full = (n0 + 16) <= N;            // tile fully in range?

    float w = nw[nc];
    if (node >= N) w = 0.0f;                     // zero OOB rows of A

    v2f a;
    if (!hi) { a.x = x[nc * 3 + 0] * w; a.y = x[nc * 3 + 1] * w; }
    else     { a.x = x[nc * 3 + 2] * w; a.y = 0.0f; }  // K=2, K=3 pad

#pragma unroll
    for (int h = 0; h < 2; ++h) {          // two 16-wide t tiles
        int t = h * 16 + tl;
        v2f bm;
        if (!hi) { bm.x = v[0 * T_DIRS + t]; bm.y = v[1 * T_DIRS + t]; }
        else     { bm.x = v[2 * T_DIRS + t]; bm.y = 0.0f;              }

        v8f c = {};
        c = __builtin_amdgcn_wmma_f32_16x16x4_f32(
                /*neg_a=*/false, a, /*neg_b=*/false, bm,
                /*c_mod=*/(short)0, c, /*reuse_a=*/false, /*reuse_b=*/false);

        int mbase = hi ? 8 : 0;
        if (full) {
#pragma unroll
            for (int r = 0; r < 8; ++r)
                nh[(n0 + mbase + r) * T_DIRS + t] = c[r];
        } else {
#pragma unroll
            for (int r = 0; r < 8; ++r) {
                int n = n0 + mbase + r;
                if (n < N) nh[n * T_DIRS + t] = c[r];
            }
        }
    }
}

// ---------------------------------------------------------------------------
// Kernel 2: eh[e, t] = max(nh[ei0[e], t], nh[ei1[e], t]) * ew[e]
//           idx_e[e] = batch[ei0[e]]
// One wave per edge; lane = t.
// ---------------------------------------------------------------------------
__global__ __launch_bounds__(256)
void eh_kernel(const float* __restrict__ nh,
               const int*  __restrict__ ei,
               const float* __restrict__ ew,
               const int*  __restrict__ batch,
               float* __restrict__ eh,
               int*  __restrict__ idx_e, int E) {
    long tid = (long)blockIdx.x * blockDim.x + threadIdx.x;
    int e = (int)(tid >> 5);
    int t = (int)(tid & 31);
    if (e >= E) return;
    int a = ei[e];
    int b = ei[(size_t)E + e];
    float ha = nh[a * T_DIRS + t];
    float hb = nh[b * T_DIRS + t];
    eh[e * T_DIRS + t] = fmaxf(ha, hb) * ew[e];
    if (t == 0) idx_e[e] = batch[a];
}

// ---------------------------------------------------------------------------
// Kernel 3: ect[b, s, t] = sum_{nodes in b} sig(SCALE*(lin[s]-nh))
//                        - sum_{edges in b} sig(SCALE*(lin[s]-eh))
// One block per graph; 8 waves x 32 lanes. Thread owns s in {w,w+8,w+16,w+24},
// t = lane -> each (s,t) cell has exactly one owner: no atomics/reductions.
// Edge graph-ids streamed through LDS with double-buffered
// GLOBAL_LOAD_ASYNC_TO_LDS_B32 (ASYNCcnt-tracked DMA, overlapped with compute).
// Every thread always issues exactly 8 async loads per chunk (indices clamped)
// so each wave's ASYNCcnt bookkeeping is exact: wait <=8 releases the
// previous chunk (async loads complete in order).
// ---------------------------------------------------------------------------
#define EDGE_CHUNK 2048

__global__ __launch_bounds__(256)
void ect_kernel(const float* __restrict__ nh,
                const float* __restrict__ eh,
                const int*  __restrict__ batch,
                const int*  __restrict__ idx_e,
                float* __restrict__ out, int N, int E) {
    const int b    = blockIdx.x;
    const int lane = threadIdx.x & 31;
    const int wv   = threadIdx.x >> 5;
    const int tidb = threadIdx.x;

    // node range for graph b via binary search on sorted batch
    int nstart, nend;
    { int l = 0, r = N;
      while (l < r) { int m = (l + r) >> 1; if (batch[m] < b) l = m + 1; else r = m; }
      nstart = l; }
    { int l = nstart, r = N;
      while (l < r) { int m = (l + r) >> 1; if (batch[m] < b + 1) l = m + 1; else r = m; }
      nend = l; }

    float linv[4], acc[4];
#pragma unroll
    for (int j = 0; j < 4; ++j) {
        int s = wv + 8 * j;
        linv[j] = -RADIUS_F + (2.0f * RADIUS_F) * (float)s / (float)(S_STEPS - 1);
        acc[j] = 0.0f;
    }

    // --- nodes ---
    for (int m = nstart; m < nend; ++m) {
        if (m + 16 < nend) __builtin_prefetch(nh + (m + 16) * T_DIRS, 0, 0);
        float h = nh[m * T_DIRS + lane];
#pragma unroll
        for (int j = 0; j < 4; ++j) {
            float z = SCALE_F * (linv[j] - h);
            acc[j] += 1.0f / (1.0f + __expf(-z));
        }
    }

    // --- edges: double-buffered async DMA of graph ids into LDS ---
    __shared__ int sg[2][EDGE_CHUNK];
    const int nchunk = (E + EDGE_CHUNK - 1) / EDGE_CHUNK;

    auto issue_chunk = [&](int c, int bufsel) {
#pragma unroll
        for (int k = 0; k < 8; ++k) {
            int i = tidb + k * 256;               // 0..2047
            int g = c * EDGE_CHUNK + i;
            if (g > E - 1) g = E - 1;             // clamp: EXEC stays all-ones
            unsigned lds_off =
                (unsigned)(unsigned long long)(&sg[bufsel][i]);
            unsigned long long gaddr =
                (unsigned long long)(const int*)(idx_e + g);
            asm volatile("global_load_async_to_lds_b32 %0, %1, off"
                         :: "v"(lds_off), "v"(gaddr)
                         : "memory");
        }
    };

    issue_chunk(0, 0);
    for (int c = 0; c < nchunk; ++c) {
        __syncthreads();                          // prev buffer free for reuse
        bool more = (c + 1) < nchunk;
        if (more) {
            issue_chunk(c + 1, (c + 1) & 1);
            asm volatile("s_wait_asynccnt 0x8" ::: "memory");   // chunk c done
        } else {
            asm volatile("s_wait_asynccnt 0x0" ::: "memory");
        }
        __syncthreads();                          // chunk c visible to all

        const int* buf = sg[c & 1];
        const int  base = c * EDGE_CHUNK;
        const int  cnt  = min(EDGE_CHUNK, E - base);
        for (int i = 0; i < cnt; ++i) {
            if (buf[i] == b) {
                float h = eh[(base + i) * T_DIRS + lane];
#pragma unroll
                for (int j = 0; j < 4; ++j) {
                    float z = SCALE_F * (linv[j] - h);
                    acc[j] -= 1.0f / (1.0f + __expf(-z));
                }
            }
        }
    }

#pragma unroll
    for (int j = 0; j < 4; ++j) {
        int s = wv + 8 * j;
        out[b * S_STEPS * T_DIRS + s * T_DIRS + lane] = acc[j];
    }
}

// ---------------------------------------------------------------------------
extern "C" void kernel_launch(void* const* d_in, const int* in_sizes, int n_in,
                              void* d_out, int out_size, void* d_ws, size_t ws_size,
                              hipStream_t stream) {
    const float* x     = (const float*)d_in[0];
    const int*   batch = (const int*)  d_in[1];
    const int*   ei    = (const int*)  d_in[2];
    const float* nw    = (const float*)d_in[3];
    const float* ew    = (const float*)d_in[4];
    const float* v     = (const float*)d_in[5];
    float* out = (float*)d_out;

    const int N = in_sizes[1];   // batch has N entries
    const int E = in_sizes[4];   // edge_weights has E entries

    // workspace: nh [N*32] f32 | eh [E*32] f32 | idx_e [E] i32  (~19.6 MB)
    float* nh    = (float*)d_ws;
    float* eh    = nh + (size_t)N * T_DIRS;
    int*   idx_e = (int*)(eh + (size_t)E * T_DIRS);

    // 1) node heights via WMMA
    int numTiles = (N + 15) >> 4;          // 16 nodes per wave
    int wavesPerBlk = 256 / 32;
    int gridA = (numTiles + wavesPerBlk - 1) / wavesPerBlk;
    nh_wmma_kernel<<<gridA, 256, 0, stream>>>(x, nw, v, nh, N);

    // 2) edge heights + edge graph ids
    long ethreads = (long)E * 32;
    int gridB = (int)((ethreads + 255) / 256);
    eh_kernel<<<gridB, 256, 0, stream>>>(nh, ei, ew, batch, eh, idx_e, E);

    // 3) ECT accumulation, one block per graph
    ect_kernel<<<N_GRAPHS, 256, 0, stream>>>(nh, eh, batch, idx_e, out, N, E);
}